// GAT_36103495090824
// MI455X (gfx1250) — compile-verified
//
#include <hip/hip_runtime.h>
#include <hip/hip_bf16.h>

// ---------------------------------------------------------------------------
// GAT forward (2 layers) for MI455X / gfx1250, wave32.
//  - feature GEMMs: v_wmma_f32_16x16x4_f32, A tile async-staged to LDS
//    (global_load_async_to_lds_b128 + s_wait_asynccnt), 32x32 patch per wave.
//  - edge softmax/aggregation: L2-resident gathers + f32 atomics.
// ---------------------------------------------------------------------------

typedef __attribute__((ext_vector_type(2))) float v2f;
typedef __attribute__((ext_vector_type(8))) float v8f;

#define N_NODES 50000
#define N_EDGES 800000
#define HEADS   4
#define FDIM    256   // HEADS * DHEAD
#define OUT_STRIDE 768
#define APAD    260   // padded LDS row stride (floats): stride%64 != 0 -> no bank conflicts

// ---------------------------------------------------------------------------
// init: out[:,0:256] = x ; out[:,256:768] = 0  (atomic-accumulation targets)
// ---------------------------------------------------------------------------
__global__ __launch_bounds__(256)
void gat_init_out(const float* __restrict__ x, float* __restrict__ out)
{
    int n = blockIdx.x;
    const float* xr = x + (size_t)n * FDIM;
    float* orow = out + (size_t)n * OUT_STRIDE;
    for (int k = 0; k < 3; ++k) {
        int c = threadIdx.x + k * 256;
        orow[c] = (c < FDIM) ? xr[c] : 0.0f;
    }
}

__global__ __launch_bounds__(256)
void gat_init_md(float* __restrict__ m, float* __restrict__ denom, int n)
{
    int i = blockIdx.x * 256 + threadIdx.x;
    if (i < n) {
        m[i]     = __int_as_float(0xFF800000); // -inf
        denom[i] = 0.0f;
    }
}

// ---------------------------------------------------------------------------
// GEMM: F[N,256] = A[N,256](row stride lda) @ W[256,256]
// grid = ceil(N/32) blocks of 256 threads (8 waves).
// Block owns a 32-row strip; wave w computes rows[0:32) x cols[32w,32w+32)
// as four 16x16 WMMA accumulator tiles. A tile staged into LDS with
// per-lane async copies (ASYNCcnt), B streamed from L2.
// ---------------------------------------------------------------------------
__global__ __launch_bounds__(256)
void gat_gemm_wmma(const float* __restrict__ A, int lda,
                   const float* __restrict__ W,
                   float* __restrict__ F, int nrows)
{
    __shared__ float As[32 * APAD];           // 33.3 KB

    const int m0 = blockIdx.x * 32;
    const int t  = threadIdx.x;

    // async-stage the 32x256 A tile (2048 x 16B chunks, 8 per thread), row-clamped
    for (int i = 0; i < 8; ++i) {
        int idx = t + i * 256;                // chunk id
        int r   = idx >> 6;                   // 64 chunks per row
        int c   = idx & 63;
        int grow = m0 + r;
        if (grow > nrows - 1) grow = nrows - 1;
        const float* gp = A + (size_t)grow * lda + c * 4;
        unsigned lds_off = (unsigned)(size_t)(&As[r * APAD + c * 4]);
        asm volatile("global_load_async_to_lds_b128 %0, %1, off"
                     :: "v"(lds_off), "v"((unsigned long long)(size_t)gp)
                     : "memory");
    }
    asm volatile("s_wait_asynccnt 0x0" ::: "memory");
    __syncthreads();

    const int lane  = t & 31;
    const int wave  = t >> 5;
    const int lhalf = lane >> 4;              // 0: lanes 0-15, 1: lanes 16-31
    const int lmod  = lane & 15;
    const int n0    = wave * 32;

    v8f c00 = {}, c01 = {}, c10 = {}, c11 = {};

    for (int k0 = 0; k0 < FDIM; k0 += 4) {
        // A fragments (16x4 layout): lanes 0-15 K=k0,k0+1 ; lanes 16-31 K=k0+2,k0+3
        const float* a0p = &As[lmod * APAD + k0 + lhalf * 2];
        const float* a1p = &As[(16 + lmod) * APAD + k0 + lhalf * 2];
        v2f a0, a1;
        a0.x = a0p[0]; a0.y = a0p[1];
        a1.x = a1p[0]; a1.y = a1p[1];

        // B fragments (4x16 layout): VGPR v -> lanes 0-15 K=k0+v, lanes 16-31 K=k0+2+v
        const float* Wk = W + (size_t)(k0 + lhalf * 2) * FDIM;
        v2f b0, b1;
        b0.x = Wk[n0 + lmod];      b0.y = Wk[FDIM + n0 + lmod];
        b1.x = Wk[n0 + 16 + lmod]; b1.y = Wk[FDIM + n0 + 16 + lmod];

        c00 = __builtin_amdgcn_wmma_f32_16x16x4_f32(false, a0, false, b0, (short)0, c00, false, false);
        c01 = __builtin_amdgcn_wmma_f32_16x16x4_f32(false, a0, false, b1, (short)0, c01, false, false);
        c10 = __builtin_amdgcn_wmma_f32_16x16x4_f32(false, a1, false, b0, (short)0, c10, false, false);
        c11 = __builtin_amdgcn_wmma_f32_16x16x4_f32(false, a1, false, b1, (short)0, c11, false, false);
    }

    // C/D layout: VGPR v -> M = v (lanes 0-15) / v+8 (lanes 16-31), N = lane%16
    for (int v = 0; v < 8; ++v) {
        int r0 = m0 + v + lhalf * 8;
        if (r0 < nrows) {
            F[(size_t)r0 * FDIM + n0 + lmod]      = c00[v];
            F[(size_t)r0 * FDIM + n0 + 16 + lmod] = c01[v];
        }
        int r1 = r0 + 16;
        if (r1 < nrows) {
            F[(size_t)r1 * FDIM + n0 + lmod]      = c10[v];
            F[(size_t)r1 * FDIM + n0 + 16 + lmod] = c11[v];
        }
    }
}

// ---------------------------------------------------------------------------
// per-node attention scores: el[n,h] = sum_d feat[n,h,d]*al[h,d] (same for er)
// ---------------------------------------------------------------------------
__global__ __launch_bounds__(256)
void gat_attn_scores(const float* __restrict__ F,
                     const float* __restrict__ al, const float* __restrict__ ar,
                     float* __restrict__ el, float* __restrict__ er)
{
    int wave = threadIdx.x >> 5;
    int lane = threadIdx.x & 31;
    int n = blockIdx.x * 8 + wave;
    if (n >= N_NODES) return;

    const float4* row = (const float4*)(F + (size_t)n * FDIM) + lane * 2;
    const float4* alv = (const float4*)al + lane * 2;
    const float4* arv = (const float4*)ar + lane * 2;

    float sl = 0.0f, sr = 0.0f;
    for (int j = 0; j < 2; ++j) {
        float4 f = row[j], l = alv[j], r = arv[j];
        sl += f.x * l.x + f.y * l.y + f.z * l.z + f.w * l.w;
        sr += f.x * r.x + f.y * r.y + f.z * r.z + f.w * r.w;
    }
    for (int off = 1; off < 8; off <<= 1) {
        sl += __shfl_xor(sl, off, 32);
        sr += __shfl_xor(sr, off, 32);
    }
    if ((lane & 7) == 0) {
        int h = lane >> 3;
        el[n * HEADS + h] = sl;
        er[n * HEADS + h] = sr;
    }
}

__device__ __forceinline__ void atomicMaxF(float* addr, float val)
{
    int iv = __float_as_int(val);
    if (iv >= 0) atomicMax((int*)addr, iv);
    else         atomicMin((unsigned int*)addr, (unsigned int)iv);
}

__global__ __launch_bounds__(256)
void gat_edge_scores(const int* __restrict__ src, const int* __restrict__ dst,
                     const float* __restrict__ el, const float* __restrict__ er,
                     float* __restrict__ e, float* __restrict__ m)
{
    int i = blockIdx.x * 256 + threadIdx.x;     // one thread per (edge, head)
    if (i >= N_EDGES * HEADS) return;
    int edge = i >> 2, h = i & 3;
    int s = src[edge], d = dst[edge];
    float v = el[s * HEADS + h] + er[d * HEADS + h];
    v = (v > 0.0f) ? v : 0.2f * v;
    e[i] = v;
    atomicMaxF(&m[d * HEADS + h], v);
}

__global__ __launch_bounds__(256)
void gat_edge_exp(const int* __restrict__ dst, const float* __restrict__ m,
                  float* __restrict__ e, float* __restrict__ denom)
{
    int i = blockIdx.x * 256 + threadIdx.x;
    if (i >= N_EDGES * HEADS) return;
    int edge = i >> 2, h = i & 3;
    int d = dst[edge];
    float ex = __expf(e[i] - m[d * HEADS + h]);
    e[i] = ex;
    atomicAdd(&denom[d * HEADS + h], ex);
}

__global__ __launch_bounds__(256)
void gat_edge_aggregate(const int* __restrict__ src, const int* __restrict__ dst,
                        const float* __restrict__ F, const float* __restrict__ e,
                        const float* __restrict__ denom, float* __restrict__ out)
{
    int wave = threadIdx.x >> 5;
    int lane = threadIdx.x & 31;
    int edge = blockIdx.x * 8 + wave;
    if (edge >= N_EDGES) return;

    int s = src[edge], d = dst[edge];
    int h = lane >> 3;                          // lane covers 8 contiguous dims
    float alpha = e[edge * HEADS + h] / denom[d * HEADS + h];

    const float4* fs = (const float4*)(F + (size_t)s * FDIM) + lane * 2;
    float4 f0 = fs[0], f1 = fs[1];

    float* od = out + (size_t)d * OUT_STRIDE + lane * 8;
    atomicAdd(od + 0, f0.x * alpha);
    atomicAdd(od + 1, f0.y * alpha);
    atomicAdd(od + 2, f0.z * alpha);
    atomicAdd(od + 3, f0.w * alpha);
    atomicAdd(od + 4, f1.x * alpha);
    atomicAdd(od + 5, f1.y * alpha);
    atomicAdd(od + 6, f1.z * alpha);
    atomicAdd(od + 7, f1.w * alpha);
}

__global__ __launch_bounds__(256)
void gat_finalize(float* __restrict__ out, const float* __restrict__ bias, int elu)
{
    int i = blockIdx.x * 256 + threadIdx.x;
    if (i >= N_NODES * FDIM) return;
    int n = i >> 8, c = i & 255;
    float* p = out + (size_t)n * OUT_STRIDE + c;
    float v = *p + bias[c];
    if (elu) v = (v > 0.0f) ? v : (__expf(v) - 1.0f);
    *p = v;
}

// ---------------------------------------------------------------------------
// launcher
// ---------------------------------------------------------------------------
extern "C" void kernel_launch(void* const* d_in, const int* in_sizes, int n_in,
                              void* d_out, int out_size, void* d_ws, size_t ws_size,
                              hipStream_t stream)
{
    const float* x   = (const float*)d_in[0];
    const int*   src = (const int*)d_in[1];
    const int*   dst = (const int*)d_in[2];
    const float* W1  = (const float*)d_in[3];
    const float* al1 = (const float*)d_in[4];
    const float* ar1 = (const float*)d_in[5];
    const float* b1  = (const float*)d_in[6];
    const float* W2  = (const float*)d_in[7];
    const float* al2 = (const float*)d_in[8];
    const float* ar2 = (const float*)d_in[9];
    const float* b2  = (const float*)d_in[10];
    float* out = (float*)d_out;

    char* ws = (char*)d_ws;
    float* feat  = (float*)(ws);                                     // 51.2 MB
    float* el    = (float*)(ws + (size_t)N_NODES * FDIM * 4);
    float* er    = el    + (size_t)N_NODES * HEADS;
    float* m     = er    + (size_t)N_NODES * HEADS;
    float* denom = m     + (size_t)N_NODES * HEADS;
    float* evals = denom + (size_t)N_NODES * HEADS;                  // 12.8 MB

    const int NH        = N_NODES * HEADS;
    const int EH        = N_EDGES * HEADS;
    const int mdBlocks  = (NH + 255) / 256;
    const int ehBlocks  = (EH + 255) / 256;
    const int aggBlocks = N_EDGES / 8;
    const int finBlocks = (N_NODES * FDIM + 255) / 256;
    const int gemmBlocks = (N_NODES + 31) / 32;                      // 1563

    gat_init_out<<<N_NODES, 256, 0, stream>>>(x, out);

    // ---------------- layer 1 ----------------
    float* h1 = out + FDIM;        // strided view, cols [256,512)
    gat_init_md<<<mdBlocks, 256, 0, stream>>>(m, denom, NH);
    gat_gemm_wmma<<<gemmBlocks, 256, 0, stream>>>(x, FDIM, W1, feat, N_NODES);
    gat_attn_scores<<<N_NODES / 8, 256, 0, stream>>>(feat, al1, ar1, el, er);
    gat_edge_scores<<<ehBlocks, 256, 0, stream>>>(src, dst, el, er, evals, m);
    gat_edge_exp<<<ehBlocks, 256, 0, stream>>>(dst, m, evals, denom);
    gat_edge_aggregate<<<aggBlocks, 256, 0, stream>>>(src, dst, feat, evals, denom, h1);
    gat_finalize<<<finBlocks, 256, 0, stream>>>(h1, b1, /*elu=*/1);

    // ---------------- layer 2 ----------------
    float* h2 = out + 2 * FDIM;    // strided view, cols [512,768)
    gat_init_md<<<mdBlocks, 256, 0, stream>>>(m, denom, NH);
    gat_gemm_wmma<<<gemmBlocks, 256, 0, stream>>>(h1, OUT_STRIDE, W2, feat, N_NODES);
    gat_attn_scores<<<N_NODES / 8, 256, 0, stream>>>(feat, al2, ar2, el, er);
    gat_edge_scores<<<ehBlocks, 256, 0, stream>>>(src, dst, el, er, evals, m);
    gat_edge_exp<<<ehBlocks, 256, 0, stream>>>(dst, m, evals, denom);
    gat_edge_aggregate<<<aggBlocks, 256, 0, stream>>>(src, dst, feat, evals, denom, h2);
    gat_finalize<<<finBlocks, 256, 0, stream>>>(h2, b2, /*elu=*/0);
}